// HebbLinear_18468359373046
// MI455X (gfx1250) — compile-verified
//
#include <hip/hip_runtime.h>

typedef float v2f __attribute__((ext_vector_type(2)));
typedef float v4f __attribute__((ext_vector_type(4)));
typedef float v8f __attribute__((ext_vector_type(8)));

#define IND  512
#define OUTD 512
#define BSZ  256

// ---------------------------------------------------------------------------
// Kernel 1: yw[b,o] = bias[o] + sum_i x[b,i] * w[i,o]
// One 16x16 output tile per wave via V_WMMA_F32_16X16X4_F32 (f32 WMMA, K=4).
// 512 tiles total (16 M-tiles x 32 N-tiles), 8 waves/block -> 64 blocks.
// ---------------------------------------------------------------------------
__global__ __launch_bounds__(256) void hebb_xw_wmma(const float* __restrict__ x,
                                                    const float* __restrict__ w,
                                                    const float* __restrict__ bias,
                                                    float* __restrict__ yw) {
  const int lane  = threadIdx.x & 31;
  const int wave  = threadIdx.x >> 5;
  const int tile  = blockIdx.x * 8 + wave;   // 0..511
  const int m0    = (tile >> 5) * 16;        // sample-tile base
  const int n0    = (tile & 31) * 16;        // out-col-tile base

  const int ml    = lane & 15;               // M (A) / N (B,C) within tile
  const int khalf = (lane >> 4) << 1;        // lanes 0-15 -> K+0/1, 16-31 -> K+2/3

  // C/D layout: VGPR r -> row m0+r (lanes 0-15) / m0+r+8 (lanes 16-31), col n0+(lane&15).
  const float bcol = bias[n0 + ml];
  v8f c;
#pragma unroll
  for (int r = 0; r < 8; ++r) c[r] = bcol;

  const float* xrow = x + (m0 + ml) * IND;
#pragma unroll 4
  for (int k = 0; k < IND; k += 4) {
    const int kk = k + khalf;
    v2f a, bm;
    a.x  = xrow[kk];                         // A[M=ml][K=kk]
    a.y  = xrow[kk + 1];                     // A[M=ml][K=kk+1]
    bm.x = w[kk * OUTD + n0 + ml];           // B[K=kk][N=ml]
    bm.y = w[(kk + 1) * OUTD + n0 + ml];     // B[K=kk+1][N=ml]
    c = __builtin_amdgcn_wmma_f32_16x16x4_f32(false, a, false, bm,
                                              (short)0, c, false, false);
  }

  const int rowBase = m0 + ((lane >> 4) << 3);
#pragma unroll
  for (int r = 0; r < 8; ++r)
    yw[(size_t)(rowBase + r) * OUTD + n0 + ml] = c[r];
}

// ---------------------------------------------------------------------------
// Kernel 2: y[b,o] = yw[b,o] + sum_i (x[b,i]*alpha[i]) * hebb[b,i,o]
// One workgroup per sample streams the 1 MB hebb[b] slab exactly once with
// non-temporal loads; xa row staged in LDS; each thread owns 2 out columns.
// ---------------------------------------------------------------------------
__global__ __launch_bounds__(256) void hebb_reduce(const float* __restrict__ x,
                                                   const float* __restrict__ hebb,
                                                   const float* __restrict__ alpha,
                                                   const float* __restrict__ yw,
                                                   float* __restrict__ yout) {
  __shared__ float xa[IND];
  const int b = blockIdx.x;
  const int t = threadIdx.x;

#pragma unroll
  for (int i = t; i < IND; i += 256)
    xa[i] = x[b * IND + i] * alpha[i];
  __syncthreads();

  float acc0 = yw[(size_t)b * OUTD + t];
  float acc1 = yw[(size_t)b * OUTD + t + 256];

  const float* hb = hebb + (size_t)b * IND * OUTD;
#pragma unroll 4
  for (int i = 0; i < IND; ++i) {
    const float xv = xa[i];
    const float* hr = hb + (size_t)i * OUTD;
    acc0 = __builtin_fmaf(xv, __builtin_nontemporal_load(hr + t), acc0);
    acc1 = __builtin_fmaf(xv, __builtin_nontemporal_load(hr + t + 256), acc1);
  }

  yout[(size_t)b * OUTD + t]       = acc0;
  yout[(size_t)b * OUTD + t + 256] = acc1;
}

// ---------------------------------------------------------------------------
// Kernel 3: delta[b,i,o] = x[b,i] * y[b,o]
// One workgroup per sample; x/y rows staged in LDS; 1 MB written as coalesced
// 128-bit non-temporal stores (write-once stream, keep it out of L2).
// ---------------------------------------------------------------------------
__global__ __launch_bounds__(256) void hebb_delta(const float* __restrict__ x,
                                                  const float* __restrict__ y,
                                                  float* __restrict__ delta) {
  __shared__ float xrow[IND];
  __shared__ float yrow[OUTD];
  const int b = blockIdx.x;
  const int t = threadIdx.x;

#pragma unroll
  for (int i = t; i < IND; i += 256)  xrow[i] = x[(size_t)b * IND + i];
#pragma unroll
  for (int o = t; o < OUTD; o += 256) yrow[o] = y[(size_t)b * OUTD + o];
  __syncthreads();

  float* db = delta + (size_t)b * IND * OUTD;
  // 512 rows x 128 float4 = 65536 vec stores; 256 threads -> 256 iterations.
#pragma unroll 2
  for (int iter = 0; iter < 256; ++iter) {
    const int idx = iter * 256 + t;
    const int i   = idx >> 7;
    const int o4  = (idx & 127) << 2;
    const float xv = xrow[i];
    const v4f yv = *reinterpret_cast<const v4f*>(&yrow[o4]);
    v4f dv;
    dv.x = xv * yv.x; dv.y = xv * yv.y; dv.z = xv * yv.z; dv.w = xv * yv.w;
    __builtin_nontemporal_store(dv,
        reinterpret_cast<v4f*>(db + (size_t)i * OUTD + o4));
  }
}

// ---------------------------------------------------------------------------
extern "C" void kernel_launch(void* const* d_in, const int* in_sizes, int n_in,
                              void* d_out, int out_size, void* d_ws, size_t ws_size,
                              hipStream_t stream) {
  const float* x     = (const float*)d_in[0];   // (1,256,512)
  const float* hebb  = (const float*)d_in[1];   // (256,512,512)
  const float* w     = (const float*)d_in[2];   // (512,512)
  const float* bias  = (const float*)d_in[3];   // (512,)
  const float* alpha = (const float*)d_in[4];   // (512,)

  float* out   = (float*)d_out;                 // y: 256*512
  float* delta = out + (size_t)BSZ * OUTD;      // delta: 256*512*512

  // yw staging (512 KB): use d_ws if big enough, else the tail of the delta
  // region (kernel 3 fully overwrites it afterwards; same-stream ordering).
  const size_t ywElems = (size_t)BSZ * OUTD;
  float* yw = (ws_size >= ywElems * sizeof(float))
                  ? (float*)d_ws
                  : (delta + (size_t)BSZ * IND * OUTD - ywElems);

  hebb_xw_wmma<<<64, 256, 0, stream>>>(x, w, bias, yw);
  hebb_reduce <<<BSZ, 256, 0, stream>>>(x, hebb, alpha, yw, out);
  hebb_delta  <<<BSZ, 256, 0, stream>>>(x, out, delta);
}